// OrthLinear_16192026705943
// MI455X (gfx1250) — compile-verified
//
#include <hip/hip_runtime.h>

typedef __attribute__((ext_vector_type(2))) float        v2f;
typedef __attribute__((ext_vector_type(8))) float        v8f;
typedef __attribute__((ext_vector_type(4))) unsigned int v4u;
typedef __attribute__((ext_vector_type(4))) int          v4i;
typedef __attribute__((ext_vector_type(8))) int          v8i;

constexpr int GROUPS = 16;
constexpr int ROWS   = 16384;   // bs (M)
constexpr int CH     = 128;     // channels == N == K

// -------------------------------------------------------------------------
// Kernel 1: build Q = H_0 H_1 ... H_127 per group (one 128-thread block/group).
// Thread t owns row t of Q entirely in VGPRs (static indices only -> no spill).
// Reflector columns staged transposed in LDS so each step reads the column as
// 32 uniform-address ds_load_b128 broadcasts.
// Q is written to scratch in WMMA B-FRAGMENT order so kernel 2's staging is a
// straight contiguous copy and each B fragment is one ds_load_b64.
//   frag element: [(kblk*8 + nt)*32 + lane]*2 + elem
//     where K = kblk*4 + (lane>>4) + 2*elem,  N = nt*16 + (lane&15)
// -------------------------------------------------------------------------
constexpr int WPAD = 132;   // 132*4 = 528 B/row: 16B aligned, breaks worst conflicts

__global__ __launch_bounds__(128) void hh_build_q(const float* __restrict__ weight,
                                                  float* __restrict__ qout) {
  __shared__ float wt[CH * WPAD];   // wt[i*WPAD + j] = weight[g][j][i]  (transposed)
  __shared__ float dinv[CH];
  const int g = blockIdx.x;
  const int t = threadIdx.x;
  const float* wg = weight + (size_t)g * CH * CH;

  // coalesced global float4 read, transposed LDS write
  for (int idx4 = t; idx4 < CH * CH / 4; idx4 += 128) {
    float4 w4 = ((const float4*)wg)[idx4];
    int flat = idx4 * 4;
    int j = flat >> 7;     // component (row of weight)
    int i = flat & 127;    // reflector index (column of weight)
    wt[(i + 0) * WPAD + j] = w4.x;
    wt[(i + 1) * WPAD + j] = w4.y;
    wt[(i + 2) * WPAD + j] = w4.z;
    wt[(i + 3) * WPAD + j] = w4.w;
  }
  __syncthreads();

  // dinv[i] = 2 / (w_i . w_i): thread t handles i = t (contiguous LDS row)
  {
    float d = 0.f;
    const float4* col = (const float4*)&wt[t * WPAD];
    #pragma unroll
    for (int j4 = 0; j4 < CH / 4; ++j4) {
      float4 w4 = col[j4];
      d += w4.x * w4.x + w4.y * w4.y + w4.z * w4.z + w4.w * w4.w;
    }
    dinv[t] = 2.0f / d;
  }
  __syncthreads();

  // Q row t in registers: Q = I, then Q <- Q * H_i sequentially
  float qr[CH];
  #pragma unroll
  for (int j = 0; j < CH; ++j) qr[j] = (j == t) ? 1.0f : 0.0f;

  #pragma unroll 1
  for (int i = 0; i < CH; ++i) {
    float wc[CH];
    const float4* col = (const float4*)&wt[i * WPAD];   // uniform addr: b128 broadcast
    #pragma unroll
    for (int j4 = 0; j4 < CH / 4; ++j4) {
      float4 w4 = col[j4];
      wc[4 * j4 + 0] = w4.x; wc[4 * j4 + 1] = w4.y;
      wc[4 * j4 + 2] = w4.z; wc[4 * j4 + 3] = w4.w;
    }
    float v = 0.f;
    #pragma unroll
    for (int j = 0; j < CH; ++j) v += qr[j] * wc[j];
    const float c = v * dinv[i];
    #pragma unroll
    for (int j = 0; j < CH; ++j) qr[j] -= c * wc[j];
  }

  // emit row K = t straight into WMMA B-fragment order (1 MB total, one-time)
  float* qf = qout + (size_t)g * CH * CH;
  const int kblk  = t >> 2;
  const int rr    = t & 3;
  const int lbase = (rr & 1) * 16;   // lane sub-base (K half selector)
  const int elem  = rr >> 1;         // which float of the lane's b64 pair
  #pragma unroll
  for (int c = 0; c < CH; ++c) {
    qf[((kblk * 8 + (c >> 4)) * 32 + lbase + (c & 15)) * 2 + elem] = qr[c];
  }
}

// -------------------------------------------------------------------------
// Kernel 2: out[g] = x[g] (16384x128) @ Q[g] (128x128), fp32 WMMA 16x16x4.
// 256 threads = 8 waves; each wave owns a 16-row M tile and all 8 N tiles.
// Q arrives pre-swizzled in fragment order -> staging is a straight 64 KB
// global->LDS copy via the Tensor Data Mover, and every B fragment is a
// single conflict-free ds_load_b64 at an immediate offset.
// -------------------------------------------------------------------------
__global__ __launch_bounds__(256) void hh_apply_q(const float* __restrict__ x,
                                                  const float* __restrict__ qfrag,
                                                  float* __restrict__ out) {
  __shared__ float Bs[CH * CH];      // 64 KB, WMMA B-fragment order
  const int g    = blockIdx.y;
  const int slab = blockIdx.x;       // 128 rows of x per block
  const int tid  = threadIdx.x;
  const int wave = tid >> 5;
  const int lane = tid & 31;
  const int nl   = lane & 15;        // N (and M) lane index within fragment
  const int kh   = lane >> 4;        // K-half selector

  const float* qg = qfrag + (size_t)g * CH * CH;

#if __has_builtin(__builtin_amdgcn_tensor_load_to_lds)
  // Tensor Data Mover: one wave issues a 1-row, 16384-element (64 KB) D#.
  if (wave == 0) {
    unsigned long long ga = (unsigned long long)(uintptr_t)qg;
    unsigned lds_base = (unsigned)(uintptr_t)(&Bs[0]);   // flat addr[31:0] = LDS offset
    v4u g0;
    g0[0] = 1u;                                          // count=1, user mode
    g0[1] = lds_base;                                    // lds_addr
    g0[2] = (unsigned)ga;                                // global_addr[31:0]
    g0[3] = (unsigned)((ga >> 32) & 0x01ffffffull)       // global_addr[56:32]
          | 0x80000000u;                                 // type=2 ("image")
    v8i g1;
    g1[0] = 0x00020000;   // workgroup_mask=0 | data_size=2 (4B)
    g1[1] = 0x40000000;   // tensor_dim0[15:0]=16384 in bits[63:48]
    g1[2] = 0x00010000;   // tensor_dim0 hi=0 | tensor_dim1=1 (bit 80)
    g1[3] = 0x40000000;   // tensor_dim1 hi=0 | tile_dim0=16384 (bits 127:112)
    g1[4] = 0x00000001;   // tile_dim1=1, tile_dim2=0
    g1[5] = 16384;        // tensor_dim0_stride[31:0]
    g1[6] = 0x40000000;   // dim0_stride hi=0 | tensor_dim1_stride[15:0]=16384
    g1[7] = 0;            // tensor_dim1_stride hi
    v4i zz4 = 0;          // groups 2/3 unused (<=2D tensor)
    v8i zz8 = 0;          // extra group (6-arg toolchain form), unused
    __builtin_amdgcn_tensor_load_to_lds(g0, g1, zz4, zz4, zz8, 0);
    __builtin_amdgcn_s_wait_tensorcnt(0);
  }
#else
  for (int i4 = tid; i4 < CH * CH / 4; i4 += 256)
    ((float4*)Bs)[i4] = ((const float4*)qg)[i4];
#endif
  __syncthreads();

  const int row0 = slab * 128 + wave * 16;
  const float* xr = x + ((size_t)g * ROWS + row0 + nl) * CH;  // this lane's A row
  const float* bp = &Bs[lane * 2];                            // fragment base

  v8f acc[8] = {};
  #pragma unroll 2
  for (int kb = 0; kb < 32; ++kb) {
    // A fragment (16x4): lane l holds x[row=l&15][4*kb + 2*(l>>4) + {0,1}]
    v2f a = *(const v2f*)(xr + kb * 4 + 2 * kh);
    #pragma unroll
    for (int nt = 0; nt < 8; ++nt) {
      // B fragment: one 8-byte LDS load per lane, immediate offset, no repack
      v2f b = *(const v2f*)(bp + (kb * 8 + nt) * 64);
      acc[nt] = __builtin_amdgcn_wmma_f32_16x16x4_f32(
          /*neg_a=*/false, a, /*neg_b=*/false, b,
          /*c_mod=*/(short)0, acc[nt], /*reuse_a=*/false, /*reuse_b=*/false);
    }
  }

  // D layout: VGPR v -> lanes 0-15 M=v, lanes 16-31 M=v+8; N = lane index
  float* og = out + ((size_t)g * ROWS + row0) * CH;
  #pragma unroll
  for (int nt = 0; nt < 8; ++nt) {
    #pragma unroll
    for (int v = 0; v < 8; ++v) {
      int row = v + kh * 8;
      og[(size_t)row * CH + nt * 16 + nl] = acc[nt][v];
    }
  }
}

extern "C" void kernel_launch(void* const* d_in, const int* in_sizes, int n_in,
                              void* d_out, int out_size, void* d_ws, size_t ws_size,
                              hipStream_t stream) {
  const float* x = (const float*)d_in[0];   // (16, 16384, 128) f32
  const float* w = (const float*)d_in[1];   // (16, 128, 128)   f32
  float* out = (float*)d_out;               // (16, 16384, 128) f32
  float* qf  = (float*)d_ws;                // scratch: fragment-order Q, 1 MB

  hh_build_q<<<GROUPS, 128, 0, stream>>>(w, qf);

  dim3 grid(ROWS / 128, GROUPS);
  hh_apply_q<<<grid, 256, 0, stream>>>(x, qf, out);
}